// MaskedCRPSEnsemble_9792525434974
// MI455X (gfx1250) — compile-verified
//
#include <hip/hip_runtime.h>
#include <hip/hip_bf16.h>
#include <stdint.h>

// ---------------- compile-time geometry ----------------
#define M_ENS            50
#define ROWS_PER_TILE    128
#define TILE_FLOATS      (ROWS_PER_TILE * M_ENS)   // 6400 floats
#define TILE_BYTES       (TILE_FLOATS * 4)         // 25600 B
#define THREADS          128                       // 4 waves (wave32)
#define CHUNKS_PER_TILE  (TILE_BYTES / 8)          // 3200 B64 chunks
#define CHUNKS_PER_THD   (CHUNKS_PER_TILE / THREADS) // 25 -> ASYNCcnt stays <= 50 < 63
#define MAX_BLOCKS       1024

typedef float v2f __attribute__((ext_vector_type(2)));
typedef float v8f __attribute__((ext_vector_type(8)));

// Async global -> LDS copy, 8 bytes per lane (GVS mode: sgpr base + vgpr byte offset).
// Tracked on ASYNCcnt. Inline asm is portable across ROCm7.2 / amdgpu-toolchain.
__device__ __forceinline__ void async_copy_b64(unsigned lds_byte, const void* base, unsigned gbyte) {
#if defined(__AMDGCN__)
  asm volatile("global_load_async_to_lds_b64 %0, %1, %2"
               :
               : "v"(lds_byte), "v"(gbyte), "s"(base)
               : "memory");
#endif
}

__device__ __forceinline__ void wait_async_le25() {
#if defined(__AMDGCN__)
  asm volatile("s_wait_asynccnt 25" ::: "memory");
#endif
}
__device__ __forceinline__ void wait_async_le0() {
#if defined(__AMDGCN__)
  asm volatile("s_wait_asynccnt 0" ::: "memory");
#endif
}

__global__ __launch_bounds__(THREADS)
void crps_main(const float* __restrict__ samples,
               const float* __restrict__ y,
               float* __restrict__ partials,
               int N, int NT, int NB, unsigned gByteClamp)
{
  __shared__ float tile[2 * TILE_FLOATS];   // 51200 B, double buffered
  __shared__ float redL[THREADS];
  __shared__ float redC[THREADS];

  const int tid = threadIdx.x;
  const int bid = blockIdx.x;

  float accL = 0.0f;   // masked loss sum for this lane
  float accC = 0.0f;   // masked row count for this lane

  if (bid < NT) {      // uniform per block
    const unsigned ldsBase = (unsigned)(uintptr_t)&tile[0];

    // ---- prologue: stage tile 'bid' into buffer 0 ----
    {
      const unsigned tbase = (unsigned)bid * (unsigned)TILE_BYTES;
#pragma unroll
      for (int i = 0; i < CHUNKS_PER_THD; ++i) {
        const unsigned ch = (unsigned)tid + (unsigned)i * (unsigned)THREADS;
        unsigned gb = tbase + ch * 8u;
        if (gb > gByteClamp) gb = gByteClamp;   // clamp (not skip): uniform issue count per wave
        async_copy_b64(ldsBase + ch * 8u, samples, gb);
      }
    }

    int buf = 0;
    for (int t = bid; t < NT; t += NB) {
      const int tn = t + NB;
      if (tn < NT) {
        // stage next tile into the other buffer while we compute this one
        const unsigned tbase = (unsigned)tn * (unsigned)TILE_BYTES;
        const unsigned dbase = ldsBase + (unsigned)(buf ^ 1) * (unsigned)TILE_BYTES;
#pragma unroll
        for (int i = 0; i < CHUNKS_PER_THD; ++i) {
          const unsigned ch = (unsigned)tid + (unsigned)i * (unsigned)THREADS;
          unsigned gb = tbase + ch * 8u;
          if (gb > gByteClamp) gb = gByteClamp;
          async_copy_b64(dbase + ch * 8u, samples, gb);
        }
        wait_async_le25();   // the 25 just issued may remain; current buffer's copies are done
      } else {
        wait_async_le0();
      }
      __syncthreads();       // current buffer visible to all waves

      // ---- compute: one row per lane, row data conflict-free in LDS (wave32, 64 banks) ----
      const int row = t * ROWS_PER_TILE + tid;
      float yv = __builtin_nanf("");
      if (row < N) yv = y[row];
      const bool valid = (yv == yv);          // row >= N => NaN => masked
      const float ysafe = valid ? yv : 0.0f;

      const float* xs = &tile[buf * TILE_FLOATS + tid * M_ENS];
      float x[M_ENS];
#pragma unroll
      for (int k = 0; k < M_ENS; ++k) x[k] = xs[k];

      float sAbs = 0.0f;
#pragma unroll
      for (int k = 0; k < M_ENS; ++k) sAbs += __builtin_fabsf(x[k] - ysafe);

      float sPair = 0.0f;   // sum_{i<j} |x_i - x_j|  ==  sorted dot with (2i-M-1)
#pragma unroll
      for (int i2 = 0; i2 < M_ENS; ++i2) {
#pragma unroll
        for (int j = i2 + 1; j < M_ENS; ++j) {
          sPair += __builtin_fabsf(x[i2] - x[j]);
        }
      }

      const float loss = sAbs * (1.0f / (float)M_ENS)
                       - sPair * (1.0f / (float)(M_ENS * (M_ENS - 1)));
      accL += valid ? loss : 0.0f;
      accC += valid ? 1.0f : 0.0f;

      __syncthreads();       // everyone done reading before this buffer is refilled
      buf ^= 1;
    }
  }

  // ---- block reduction of (accL, accC); EXEC is all-ones here (no divergence) ----
  const int lane = tid & 31;
  const int wv   = tid >> 5;
  int nred;

#if defined(__AMDGCN__) && __has_builtin(__builtin_amdgcn_wmma_f32_16x16x4_f32)
  // A(16x4) x ones(4x16): lane m (and m+16) contribute A[m,*]; with a[1]=0 the layout
  // guarantees D[m,n] = accL_m + accL_{m+16} regardless of K-slot ordering.
  v2f a; a[0] = accL; a[1] = 0.0f;
  v2f b; b[0] = 1.0f; b[1] = 1.0f;
  v8f c = {0.0f,0.0f,0.0f,0.0f,0.0f,0.0f,0.0f,0.0f};
  v8f dL = __builtin_amdgcn_wmma_f32_16x16x4_f32(false, a, false, b, (short)0, c, false, false);
  a[0] = accC;
  v8f dC = __builtin_amdgcn_wmma_f32_16x16x4_f32(false, a, false, b, (short)0, c, false, false);
  // Lanes 0..15 hold D rows 0..7, lanes 16..31 hold rows 8..15; each half-sum covers 16 lanes.
  const float hL = ((dL[0]+dL[1])+(dL[2]+dL[3])) + ((dL[4]+dL[5])+(dL[6]+dL[7]));
  const float hC = ((dC[0]+dC[1])+(dC[2]+dC[3])) + ((dC[4]+dC[5])+(dC[6]+dC[7]));
  if (lane == 0 || lane == 16) {
    redL[wv * 2 + (lane >> 4)] = hL;
    redC[wv * 2 + (lane >> 4)] = hC;
  }
  nred = 2 * (THREADS / 32);
#else
  redL[tid] = accL;
  redC[tid] = accC;
  nred = THREADS;
#endif
  __syncthreads();

  if (tid == 0) {            // fixed-order, deterministic
    float s = 0.0f, cc = 0.0f;
    for (int i = 0; i < nred; ++i) { s += redL[i]; cc += redC[i]; }
    partials[2 * bid]     = s;
    partials[2 * bid + 1] = cc;
  }
}

__global__ void crps_finalize(const float* __restrict__ partials,
                              float* __restrict__ out, int nb)
{
  if (blockIdx.x == 0 && threadIdx.x == 0) {
    float s = 0.0f, c = 0.0f;
    for (int i = 0; i < nb; ++i) { s += partials[2 * i]; c += partials[2 * i + 1]; }
    out[0] = s / c;
  }
}

extern "C" void kernel_launch(void* const* d_in, const int* in_sizes, int n_in,
                              void* d_out, int out_size, void* d_ws, size_t ws_size,
                              hipStream_t stream) {
  const float* samples = (const float*)d_in[0];   // (N, M, 1) flat, N*M floats
  const float* y       = (const float*)d_in[1];   // (N, 1) flat, N floats
  float* out           = (float*)d_out;           // 1 float
  float* partials      = (float*)d_ws;

  const int NM = in_sizes[0];
  const int N  = in_sizes[1];
  const int NT = (N + ROWS_PER_TILE - 1) / ROWS_PER_TILE;

  int NB = MAX_BLOCKS;
  if (NB > NT) NB = NT;
  const size_t need = (size_t)NB * 2 * sizeof(float);
  if (need > ws_size) NB = (int)(ws_size / (2 * sizeof(float)));
  if (NB < 1) NB = 1;

  const unsigned gByteClamp = (unsigned)NM * 4u - 8u;  // last valid B64 source byte

  crps_main<<<NB, THREADS, 0, stream>>>(samples, y, partials, N, NT, NB, gByteClamp);
  crps_finalize<<<1, 32, 0, stream>>>(partials, out, NB);
}